// ANI_51049981280463
// MI455X (gfx1250) — compile-verified
//
#include <hip/hip_runtime.h>
#include <math.h>

#define B_    16
#define N_    256
#define NN_   48
#define NT_   512
#define NRAD_ 16
#define NANG_ 8
#define RC_   6.0f
#define PI_F  3.14159265358979323846f

typedef float v2f __attribute__((ext_vector_type(2)));
typedef float v8f __attribute__((ext_vector_type(8)));

// ---------------------------------------------------------------------------
// Kernel 1: pos = positions @ scaling  [B,N,3];  cel = cell @ scaling [B,3,3]
// ---------------------------------------------------------------------------
__global__ void ani_prep(const float* __restrict__ positions,
                         const float* __restrict__ scaling,
                         const float* __restrict__ cell,
                         float* __restrict__ pos,
                         float* __restrict__ cel) {
    __shared__ float sS[9];
    const int b = blockIdx.x;
    const int t = threadIdx.x;           // one thread per atom, blockDim = N_
    if (t < 9) sS[t] = scaling[b * 9 + t];
    __syncthreads();
    const float p0 = positions[(b * N_ + t) * 3 + 0];
    const float p1 = positions[(b * N_ + t) * 3 + 1];
    const float p2 = positions[(b * N_ + t) * 3 + 2];
#pragma unroll
    for (int d = 0; d < 3; ++d)
        pos[(b * N_ + t) * 3 + d] = p0 * sS[0 * 3 + d] + p1 * sS[1 * 3 + d] + p2 * sS[2 * 3 + d];
    if (t < 9) {
        const int r = t / 3, d = t % 3;
        cel[b * 9 + t] = cell[b * 9 + r * 3 + 0] * sS[0 * 3 + d] +
                         cell[b * 9 + r * 3 + 1] * sS[1 * 3 + d] +
                         cell[b * 9 + r * 3 + 2] * sS[2 * 3 + d];
    }
}

__device__ __forceinline__ float cutoff_fn(float r) {
    return (r < RC_) ? 0.5f * (__cosf(r * (PI_F / RC_)) + 1.0f) : 0.0f;
}

// ---------------------------------------------------------------------------
// Kernel 2: radial BehlerG1 -> rep[atom][0..15]. One wave per atom.
// ---------------------------------------------------------------------------
__global__ void __launch_bounds__(256)
ani_radial(const float* __restrict__ pos, const float* __restrict__ cel,
           const float* __restrict__ offsets, const float* __restrict__ mask,
           const int* __restrict__ neighbors,
           const float* __restrict__ etas_rad, const float* __restrict__ rss,
           float* __restrict__ rep) {
    __shared__ float sPos[N_ * 3];
    __shared__ float sCel[9];
    __shared__ float sEta[NRAD_], sRs[NRAD_];
    const int tid = threadIdx.x;
    const int b  = blockIdx.x >> 5;            // 32 chunks of 8 atoms each
    const int n0 = (blockIdx.x & 31) * 8;
    for (int i = tid; i < N_ * 3; i += 256) sPos[i] = pos[b * N_ * 3 + i];
    if (tid < 9)     sCel[tid] = cel[b * 9 + tid];
    if (tid < NRAD_) { sEta[tid] = etas_rad[tid]; sRs[tid] = rss[tid]; }
    __syncthreads();

    const int wave = tid >> 5, lane = tid & 31;
    const int n = n0 + wave;
    const float xi = sPos[n * 3 + 0], yi = sPos[n * 3 + 1], zi = sPos[n * 3 + 2];

    float acc[NRAD_];
#pragma unroll
    for (int e = 0; e < NRAD_; ++e) acc[e] = 0.0f;

    const int baseNM = (b * N_ + n) * NN_;
    for (int m = lane; m < NN_; m += 32) {
        const int j = neighbors[baseNM + m];
        const float o0 = offsets[(baseNM + m) * 3 + 0];
        const float o1 = offsets[(baseNM + m) * 3 + 1];
        const float o2 = offsets[(baseNM + m) * 3 + 2];
        const float dx = sPos[j * 3 + 0] + o0 * sCel[0] + o1 * sCel[3] + o2 * sCel[6] - xi;
        const float dy = sPos[j * 3 + 1] + o0 * sCel[1] + o1 * sCel[4] + o2 * sCel[7] - yi;
        const float dz = sPos[j * 3 + 2] + o0 * sCel[2] + o1 * sCel[5] + o2 * sCel[8] - zi;
        const float r = sqrtf(dx * dx + dy * dy + dz * dz);
        const float w = cutoff_fn(r) * mask[baseNM + m];
#pragma unroll
        for (int e = 0; e < NRAD_; ++e) {
            const float d = r - sRs[e];
            acc[e] += __expf(-sEta[e] * d * d) * w;
        }
    }
#pragma unroll
    for (int e = 0; e < NRAD_; ++e)
        for (int o = 16; o > 0; o >>= 1) acc[e] += __shfl_xor(acc[e], o, 32);
    if (lane == 0) {
        const int rb = (b * N_ + n) * 32;
#pragma unroll
        for (int e = 0; e < NRAD_; ++e) rep[rb + e] = acc[e];
    }
}

// ---------------------------------------------------------------------------
// Kernel 3: angular BehlerG3 -> rep[atom][16..31]. One block per atom.
// (zeta == 1 => ang = [base, 4*base], so only 8 accumulators needed)
// ---------------------------------------------------------------------------
__global__ void __launch_bounds__(256)
ani_angular(const float* __restrict__ pos, const float* __restrict__ cel,
            const int* __restrict__ nj, const int* __restrict__ nk,
            const float* __restrict__ offj, const float* __restrict__ offk,
            const float* __restrict__ maskt, const float* __restrict__ etas_ang,
            float* __restrict__ rep) {
    __shared__ float sPos[N_ * 3];
    __shared__ float sCel[9];
    __shared__ float sEta[NANG_];
    __shared__ float sRed[8 * NANG_];
    const int tid  = threadIdx.x;
    const int atom = blockIdx.x;          // b*N + n
    const int b    = atom >> 8;
    const int n    = atom & 255;
    for (int i = tid; i < N_ * 3; i += 256) sPos[i] = pos[b * N_ * 3 + i];
    if (tid < 9)     sCel[tid] = cel[b * 9 + tid];
    if (tid < NANG_) sEta[tid] = etas_ang[tid];
    __syncthreads();

    const float xi = sPos[n * 3 + 0], yi = sPos[n * 3 + 1], zi = sPos[n * 3 + 2];
    float acc[NANG_];
#pragma unroll
    for (int e = 0; e < NANG_; ++e) acc[e] = 0.0f;

    const int base = atom * NT_;
    for (int s = tid; s < NT_; s += 256) {
        if (s + 256 < NT_) {  // stream-ahead hint -> global_prefetch_b8
            __builtin_prefetch(offj + (base + s + 256) * 3, 0, 0);
            __builtin_prefetch(offk + (base + s + 256) * 3, 0, 0);
        }
        const int j = nj[base + s];
        const int k = nk[base + s];
        const float a0 = offj[(base + s) * 3 + 0], a1 = offj[(base + s) * 3 + 1], a2 = offj[(base + s) * 3 + 2];
        const float c0 = offk[(base + s) * 3 + 0], c1 = offk[(base + s) * 3 + 1], c2 = offk[(base + s) * 3 + 2];
        const float pjx = sPos[j * 3 + 0] + a0 * sCel[0] + a1 * sCel[3] + a2 * sCel[6];
        const float pjy = sPos[j * 3 + 1] + a0 * sCel[1] + a1 * sCel[4] + a2 * sCel[7];
        const float pjz = sPos[j * 3 + 2] + a0 * sCel[2] + a1 * sCel[5] + a2 * sCel[8];
        const float pkx = sPos[k * 3 + 0] + c0 * sCel[0] + c1 * sCel[3] + c2 * sCel[6];
        const float pky = sPos[k * 3 + 1] + c0 * sCel[1] + c1 * sCel[4] + c2 * sCel[7];
        const float pkz = sPos[k * 3 + 2] + c0 * sCel[2] + c1 * sCel[5] + c2 * sCel[8];
        const float djx = pjx - xi, djy = pjy - yi, djz = pjz - zi;
        const float dkx = pkx - xi, dky = pky - yi, dkz = pkz - zi;
        const float dex = pkx - pjx, dey = pky - pjy, dez = pkz - pjz;
        const float rij2 = djx * djx + djy * djy + djz * djz;
        const float rik2 = dkx * dkx + dky * dky + dkz * dkz;
        const float rjk2 = dex * dex + dey * dey + dez * dez;
        const float rij = sqrtf(rij2), rik = sqrtf(rik2);
        const float mval = (maskt[base + s] > 0.0f) ? 1.0f : 0.0f;
        const float w = cutoff_fn(rij) * cutoff_fn(rik) * mval;
        const float denom = fmaxf(2.0f * rij * rik, 1e-20f);
        const float cost = mval * (rij2 + rik2 - rjk2) / denom;
        const float bw = (1.0f - cost) * w;   // base * cut(ij)*cut(ik)*mask
        const float q = -(rij2 + rik2);
#pragma unroll
        for (int e = 0; e < NANG_; ++e) acc[e] += __expf(sEta[e] * q) * bw;
    }
#pragma unroll
    for (int e = 0; e < NANG_; ++e)
        for (int o = 16; o > 0; o >>= 1) acc[e] += __shfl_xor(acc[e], o, 32);
    const int wave = tid >> 5, lane = tid & 31;
    if (lane == 0) {
#pragma unroll
        for (int e = 0; e < NANG_; ++e) sRed[wave * NANG_ + e] = acc[e];
    }
    __syncthreads();
    if (tid < NANG_) {
        float S = 0.0f;
#pragma unroll
        for (int w = 0; w < 8; ++w) S += sRed[w * NANG_ + tid];
        const int rb = atom * 32 + 16;
        rep[rb + tid * 2 + 0] = S;          // 2^(1-zeta) * base, zeta=1
        rep[rb + tid * 2 + 1] = 4.0f * S;   // 2^(1+zeta) * base
    }
}

// ---------------------------------------------------------------------------
// Kernel 4: energy[b] = sum_n rep[b,n,:] . W + N*bias  via V_WMMA_F32_16X16X4_F32
// A[16x4] = rep tile (ISA layout: lanes 0-15 -> K=0,1; lanes 16-31 -> K=2,3),
// B[4x16] = W[k] broadcast to all 16 columns -> every D column = per-atom dot.
// Uniform control flow => EXEC all ones (WMMA requirement).
// ---------------------------------------------------------------------------
__global__ void __launch_bounds__(256)
ani_energy(const float* __restrict__ rep, const float* __restrict__ W,
           const float* __restrict__ bias, float* __restrict__ out) {
    __shared__ float sRed[8];
    const int b = blockIdx.x;
    const int tid = threadIdx.x;
    const int wave = tid >> 5, lane = tid & 31;
    const int m = lane & 15, half = lane >> 4;

    v8f c = {0.f, 0.f, 0.f, 0.f, 0.f, 0.f, 0.f, 0.f};
    for (int tile = wave; tile < 16; tile += 8) {   // 2 tiles per wave, uniform
        const int atom = b * N_ + tile * 16 + m;
        const float* ra = rep + atom * 32;
#pragma unroll
        for (int kk = 0; kk < 8; ++kk) {            // K = 32 in steps of 4
            const int kb = kk * 4 + 2 * half;
            v2f a, bw;
            a.x  = ra[kb];     a.y  = ra[kb + 1];
            bw.x = W[kb];      bw.y = W[kb + 1];
            c = __builtin_amdgcn_wmma_f32_16x16x4_f32(
                    false, a, false, bw, (short)0, c, false, false);
        }
    }
    // All 16 D columns are identical; summing all lanes/VGPRs = 16 * tile-sum.
    float s = c[0] + c[1] + c[2] + c[3] + c[4] + c[5] + c[6] + c[7];
    for (int o = 16; o > 0; o >>= 1) s += __shfl_xor(s, o, 32);
    if (lane == 0) sRed[wave] = s;
    __syncthreads();
    if (tid == 0) {
        float t = 0.0f;
#pragma unroll
        for (int w = 0; w < 8; ++w) t += sRed[w];
        out[b] = t * (1.0f / 16.0f) + (float)N_ * bias[0];
    }
}

// ---------------------------------------------------------------------------
extern "C" void kernel_launch(void* const* d_in, const int* in_sizes, int n_in,
                              void* d_out, int out_size, void* d_ws, size_t ws_size,
                              hipStream_t stream) {
    (void)in_sizes; (void)n_in; (void)out_size; (void)ws_size;
    const float* positions = (const float*)d_in[0];
    const float* scaling   = (const float*)d_in[1];
    const float* cell      = (const float*)d_in[2];
    const float* offsets   = (const float*)d_in[3];
    const float* mask      = (const float*)d_in[4];
    const int*   neighbors = (const int*)  d_in[5];
    const int*   nj        = (const int*)  d_in[6];
    const int*   nk        = (const int*)  d_in[7];
    const float* offj      = (const float*)d_in[8];
    const float* offk      = (const float*)d_in[9];
    const float* maskt     = (const float*)d_in[10];
    const float* etas_rad  = (const float*)d_in[11];
    const float* rss       = (const float*)d_in[12];
    const float* etas_ang  = (const float*)d_in[13];
    const float* W         = (const float*)d_in[14];
    const float* bias      = (const float*)d_in[15];

    float* ws  = (float*)d_ws;
    float* pos = ws;                         // B*N*3   = 12288 floats
    float* cel = pos + B_ * N_ * 3;          // B*9     =   144 floats
    float* rep = cel + B_ * 9;               // B*N*32  = 131072 floats

    ani_prep   <<<B_,        N_,  0, stream>>>(positions, scaling, cell, pos, cel);
    ani_radial <<<B_ * N_ / 8, 256, 0, stream>>>(pos, cel, offsets, mask, neighbors,
                                                 etas_rad, rss, rep);
    ani_angular<<<B_ * N_,   256, 0, stream>>>(pos, cel, nj, nk, offj, offk,
                                               maskt, etas_ang, rep);
    ani_energy <<<B_,        256, 0, stream>>>(rep, W, bias, (float*)d_out);
}